// VerticalGNN_22376779612624
// MI455X (gfx1250) — compile-verified
//
#include <hip/hip_runtime.h>

// ---------------------------------------------------------------------------
// Problem constants (fixed by the reference)
// ---------------------------------------------------------------------------
#define NN_ 50000
#define EE_ 500000
#define FF_ 128
#define EDD_ 16
#define HH_ 4
#define DD_ 32
#define HDIM_ 128   // H*D
#define BB_ 256

typedef __attribute__((ext_vector_type(2))) float v2f;
typedef __attribute__((ext_vector_type(8))) float v8f;

// ---------------------------------------------------------------------------
// Fast f32 GEMM + bias (+optional ReLU) via V_WMMA_F32_16X16X4_F32.
// REQUIRES: M % 16 == 0, K % 16 == 0, Nc % 32 == 0   (true for all uses here).
// One wave computes a 16x32 tile (two 16x16 accumulators sharing the A load).
// A (16x4 f32):  lane l<16 holds row (tm+l), K = k+0/k+1 in vgpr0/1;
//                lane l>=16 holds row (tm+l-16), K = k+2/k+3.
// B (4x16 f32):  lanes 0-15 hold K=k+0 (v0) / k+1 (v1), col tn+lane;
//                lanes 16-31 hold K=k+2 / k+3.
// C/D (16x16):   vgpr r: lanes 0-15 -> row tm+r, lanes 16-31 -> row tm+r+8.
// ---------------------------------------------------------------------------
__global__ __launch_bounds__(256) void gemm_wmma_f32(
    const float* __restrict__ X, const float* __restrict__ W,
    const float* __restrict__ bias, float* __restrict__ Y,
    int M, int K, int Nc, int relu)
{
    const int lane = threadIdx.x & 31;
    const int wave = threadIdx.x >> 5;
    const int tilesN2 = Nc >> 5;                       // 16x32 tile pairs along N
    const int tile = blockIdx.x * (blockDim.x >> 5) + wave;
    if (tile >= (M >> 4) * tilesN2) return;            // wave-uniform exit
    const int tm = (tile / tilesN2) << 4;
    const int tn = (tile % tilesN2) << 5;
    const int half = lane >> 4;
    const int l16  = lane & 15;

    const float* xp  = X + (size_t)(tm + l16) * K + (half << 1);
    const float* wp0 = W + (size_t)(half << 1) * Nc + tn + l16;   // col tn+l16
    const float* wp1 = wp0 + 16;                                  // col tn+16+l16

    v8f acc0 = {0.f, 0.f, 0.f, 0.f, 0.f, 0.f, 0.f, 0.f};
    v8f acc1 = {0.f, 0.f, 0.f, 0.f, 0.f, 0.f, 0.f, 0.f};

    for (int k = 0; k < K; k += 16) {
#pragma unroll
        for (int kk = 0; kk < 16; kk += 4) {
            const v2f a = *(const v2f*)(xp + k + kk);     // 8B contiguous load
            const size_t off = (size_t)(k + kk) * Nc;
            v2f b0, b1;
            b0.x = wp0[off]; b0.y = wp0[off + Nc];
            b1.x = wp1[off]; b1.y = wp1[off + Nc];
            acc0 = __builtin_amdgcn_wmma_f32_16x16x4_f32(
                false, a, false, b0, (short)0, acc0, false, false);
            acc1 = __builtin_amdgcn_wmma_f32_16x16x4_f32(
                false, a, false, b1, (short)0, acc1, false, false);
        }
    }

    const float bv0 = bias[tn + l16];
    const float bv1 = bias[tn + 16 + l16];
    float* yp = Y + (size_t)(tm + (half << 3)) * Nc + tn + l16;
#pragma unroll
    for (int r = 0; r < 8; ++r) {
        float v0 = acc0[r] + bv0;
        float v1 = acc1[r] + bv1;
        if (relu) { v0 = fmaxf(v0, 0.f); v1 = fmaxf(v1, 0.f); }
        yp[(size_t)r * Nc]      = v0;
        yp[(size_t)r * Nc + 16] = v1;
    }
}

// ---------------------------------------------------------------------------
// Tiny final projection: y[m] = X[m,:K] . w + b   (one wave per row)
// ---------------------------------------------------------------------------
__global__ void rowvec_dot(const float* __restrict__ X, const float* __restrict__ w,
                           const float* __restrict__ b, float* __restrict__ y,
                           int M, int K)
{
    const int row  = blockIdx.x * (blockDim.x >> 5) + (threadIdx.x >> 5);
    const int lane = threadIdx.x & 31;
    if (row >= M) return;
    float p = 0.f;
    for (int c = lane; c < K; c += 32) p += X[(size_t)row * K + c] * w[c];
#pragma unroll
    for (int off = 16; off > 0; off >>= 1) p += __shfl_xor(p, off, 32);
    if (lane == 0) y[row] = p + b[0];
}

// ---------------------------------------------------------------------------
// Utility fills / copies (graph-capture-safe zero init)
// ---------------------------------------------------------------------------
__global__ void fill_f32(float* __restrict__ p, float v, size_t n)
{
    size_t i = (size_t)blockIdx.x * blockDim.x + threadIdx.x;
    if (i < n) p[i] = v;
}
__global__ void fill_u32(unsigned* __restrict__ p, unsigned v, size_t n)
{
    size_t i = (size_t)blockIdx.x * blockDim.x + threadIdx.x;
    if (i < n) p[i] = v;
}
__global__ void copy_f32(float* __restrict__ d, const float* __restrict__ s, size_t n)
{
    size_t i = (size_t)blockIdx.x * blockDim.x + threadIdx.x;
    if (i < n) d[i] = s[i];
}

// ---------------------------------------------------------------------------
// Attention pass 1: per-(edge,head) logits + encoded atomic segment-max.
// e_hd = edge_attr[e] . We[:, h*32+d], recomputed from LDS-cached We (16x128).
// ---------------------------------------------------------------------------
__device__ __forceinline__ unsigned enc_float(float x)
{
    unsigned u = __float_as_uint(x);
    return (u & 0x80000000u) ? ~u : (u | 0x80000000u);
}

__device__ __forceinline__ void load_attr16(const float* __restrict__ ea, int e,
                                            float attr[EDD_])
{
    const float4* p = (const float4*)(ea + (size_t)e * EDD_);  // 64B aligned
    float4 a0 = p[0], a1 = p[1], a2 = p[2], a3 = p[3];
    attr[0] = a0.x;  attr[1] = a0.y;  attr[2] = a0.z;  attr[3] = a0.w;
    attr[4] = a1.x;  attr[5] = a1.y;  attr[6] = a1.z;  attr[7] = a1.w;
    attr[8] = a2.x;  attr[9] = a2.y;  attr[10] = a2.z; attr[11] = a2.w;
    attr[12] = a3.x; attr[13] = a3.y; attr[14] = a3.z; attr[15] = a3.w;
}

__global__ void edge_logits(const float* __restrict__ q,
                            const float* __restrict__ kf,
                            const float* __restrict__ ea,
                            const float* __restrict__ We,
                            const int* __restrict__ src,
                            const int* __restrict__ dst,
                            float* __restrict__ logits,
                            unsigned* __restrict__ menc,
                            int E)
{
    __shared__ float sWe[EDD_ * HDIM_];
    for (int i = threadIdx.x; i < EDD_ * HDIM_; i += blockDim.x) sWe[i] = We[i];
    __syncthreads();

    int t = blockIdx.x * blockDim.x + threadIdx.x;
    if (t >= E * HH_) return;
    const int e = t >> 2, h = t & 3;
    const int s = src[e], d = dst[e];

    float attr[EDD_];
    load_attr16(ea, e, attr);

    const float* qp = q  + (size_t)d * HDIM_ + h * DD_;
    const float* kp = kf + (size_t)s * HDIM_ + h * DD_;
    float acc = 0.f;
#pragma unroll 4
    for (int dd = 0; dd < DD_; ++dd) {
        float ev = 0.f;
#pragma unroll
        for (int i = 0; i < EDD_; ++i) ev += attr[i] * sWe[i * HDIM_ + h * DD_ + dd];
        acc += qp[dd] * (kp[dd] + ev);
    }
    acc *= 0.17677669529663687f;  // 1/sqrt(D)
    logits[t] = acc;
    atomicMax(menc + (size_t)d * HH_ + h, enc_float(acc));
}

// ---------------------------------------------------------------------------
// Attention pass 2: w = exp(logit - max); atomic segment-sum of w.
// ---------------------------------------------------------------------------
__global__ void edge_softmax(float* __restrict__ wbuf,
                             const unsigned* __restrict__ menc,
                             const int* __restrict__ dst,
                             float* __restrict__ sbuf,
                             int E)
{
    int t = blockIdx.x * blockDim.x + threadIdx.x;
    if (t >= E * HH_) return;
    const int e = t >> 2, h = t & 3;
    const int d = dst[e];
    unsigned u = menc[(size_t)d * HH_ + h];
    float m;
    if (u == 0u) {
        m = 0.f;  // no arrivals: reference maps non-finite max to 0
    } else {
        unsigned v = (u & 0x80000000u) ? (u & 0x7FFFFFFFu) : ~u;
        m = __uint_as_float(v);
    }
    float w = __expf(wbuf[t] - m);
    wbuf[t] = w;
    atomicAdd(sbuf + (size_t)d * HH_ + h, w);
}

// ---------------------------------------------------------------------------
// Attention pass 3: msg = (v[src] + e) * (w / (s[dst]+eps)); scatter-add.
// ---------------------------------------------------------------------------
__global__ void edge_message(const float* __restrict__ vf,
                             const float* __restrict__ ea,
                             const float* __restrict__ We,
                             const float* __restrict__ wbuf,
                             const float* __restrict__ sbuf,
                             const int* __restrict__ src,
                             const int* __restrict__ dst,
                             float* __restrict__ outb,
                             int E)
{
    __shared__ float sWe[EDD_ * HDIM_];
    for (int i = threadIdx.x; i < EDD_ * HDIM_; i += blockDim.x) sWe[i] = We[i];
    __syncthreads();

    int t = blockIdx.x * blockDim.x + threadIdx.x;
    if (t >= E * HH_) return;
    const int e = t >> 2, h = t & 3;
    const int s = src[e], d = dst[e];
    const float a = wbuf[t] / (sbuf[(size_t)d * HH_ + h] + 1e-16f);

    float attr[EDD_];
    load_attr16(ea, e, attr);

    const float* vp = vf + (size_t)s * HDIM_ + h * DD_;
    float* op = outb + (size_t)d * HDIM_ + h * DD_;
#pragma unroll 4
    for (int dd = 0; dd < DD_; ++dd) {
        float ev = 0.f;
#pragma unroll
        for (int i = 0; i < EDD_; ++i) ev += attr[i] * sWe[i * HDIM_ + h * DD_ + dd];
        atomicAdd(op + dd, (vp[dd] + ev) * a);
    }
}

// ---------------------------------------------------------------------------
// Beta gate: one wave per node; 384-wide dot reduced with wave32 shuffles.
// ---------------------------------------------------------------------------
__global__ void beta_combine(const float* __restrict__ outb,
                             const float* __restrict__ skp,
                             const float* __restrict__ Wb,
                             float* __restrict__ hout,
                             int N)
{
    const int node = blockIdx.x * (blockDim.x >> 5) + (threadIdx.x >> 5);
    const int lane = threadIdx.x & 31;
    if (node >= N) return;

    float o[4], sk[4];
    float partial = 0.f;
#pragma unroll
    for (int j = 0; j < 4; ++j) {
        const int c = lane + 32 * j;
        o[j]  = outb[(size_t)node * HDIM_ + c];
        sk[j] = skp [(size_t)node * HDIM_ + c];
        partial += o[j] * Wb[c] + sk[j] * Wb[HDIM_ + c] + (o[j] - sk[j]) * Wb[2 * HDIM_ + c];
    }
#pragma unroll
    for (int off = 16; off > 0; off >>= 1) partial += __shfl_xor(partial, off, 32);
    const float beta = 1.f / (1.f + __expf(-partial));
#pragma unroll
    for (int j = 0; j < 4; ++j) {
        const int c = lane + 32 * j;
        hout[(size_t)node * HDIM_ + c] = beta * sk[j] + (1.f - beta) * o[j];
    }
}

// ---------------------------------------------------------------------------
// Segment pooling over nodes: acc[batch[n], c] += h[n, c]; optional counts.
// ---------------------------------------------------------------------------
__global__ void pool_accum(const float* __restrict__ h,
                           const int* __restrict__ bidx,
                           float* __restrict__ acc,
                           float* __restrict__ counts,
                           int N, int shift)
{
    size_t t = (size_t)blockIdx.x * blockDim.x + threadIdx.x;
    const int C = 1 << shift;
    if (t >= ((size_t)N << shift)) return;
    const int node = (int)(t >> shift);
    const int c = (int)(t & (C - 1));
    const int b = bidx[node];
    atomicAdd(acc + (size_t)b * C + c, h[(size_t)node * C + c]);
    if (counts && c == 0) atomicAdd(counts + b, 1.f);
}

// ---------------------------------------------------------------------------
// GIN neighborhood scatter: agg[dst, c] += h[src, c].
// ---------------------------------------------------------------------------
__global__ void gin_scatter(const float* __restrict__ h,
                            const int* __restrict__ src,
                            const int* __restrict__ dst,
                            float* __restrict__ agg,
                            int E, int shift)
{
    size_t t = (size_t)blockIdx.x * blockDim.x + threadIdx.x;
    const int C = 1 << shift;
    if (t >= ((size_t)E << shift)) return;
    const int e = (int)(t >> shift);
    const int c = (int)(t & (C - 1));
    atomicAdd(agg + (size_t)dst[e] * C + c, h[(size_t)src[e] * C + c]);
}

// ---------------------------------------------------------------------------
// z = concat(mean_pool(gt/counts), add_pool(gin))  -> [B, 160]
// ---------------------------------------------------------------------------
__global__ void assemble_z(const float* __restrict__ gt,
                           const float* __restrict__ counts,
                           const float* __restrict__ gp,
                           float* __restrict__ z,
                           int B)
{
    int t = blockIdx.x * blockDim.x + threadIdx.x;
    if (t >= B * (HDIM_ + DD_)) return;
    const int b = t / (HDIM_ + DD_);
    const int c = t % (HDIM_ + DD_);
    z[t] = (c < HDIM_) ? gt[(size_t)b * HDIM_ + c] / fmaxf(counts[b], 1.f)
                       : gp[(size_t)b * DD_ + (c - HDIM_)];
}

// ---------------------------------------------------------------------------
// Host orchestration
// ---------------------------------------------------------------------------
extern "C" void kernel_launch(void* const* d_in, const int* in_sizes, int n_in,
                              void* d_out, int out_size, void* d_ws, size_t ws_size,
                              hipStream_t stream)
{
    auto F = [&](int i) { return (const float*)d_in[i]; };
    auto I = [&](int i) { return (const int*)d_in[i]; };

    const float *x = nullptr, *ea = nullptr;
    const int *eidx = nullptr, *bidx = nullptr;
    const float *Wq[2], *bq[2], *Wk[2], *bk[2], *Wv[2], *bv[2];
    const float *We[2], *Ws[2], *bs[2], *Wb[2];
    const float *gW1[2], *gb1[2], *gW2[2], *gb2[2];
    const float *roW1, *rob1, *roW2, *rob2;

    auto load_sorted_params = [&](int p) {
        for (int L = 0; L < 2; ++L) {
            gW1[L] = F(p); gW2[L] = F(p + 1); gb1[L] = F(p + 2); gb2[L] = F(p + 3);
            p += 4;
        }
        roW1 = F(p); roW2 = F(p + 1); rob1 = F(p + 2); rob2 = F(p + 3); p += 4;
        for (int L = 0; L < 2; ++L) {
            Wb[L] = F(p);     We[L] = F(p + 1); Wk[L] = F(p + 2); Wq[L] = F(p + 3);
            Ws[L] = F(p + 4); Wv[L] = F(p + 5); bk[L] = F(p + 6); bq[L] = F(p + 7);
            bs[L] = F(p + 8); bv[L] = F(p + 9);
            p += 10;
        }
        return p;
    };

    if (in_sizes[0] == NN_ * FF_) {
        // top-level insertion order: x, edge_attr, edge_index, batch_index, params...
        x = F(0); ea = F(1); eidx = I(2); bidx = I(3);
        if (in_sizes[4] == FF_ * HDIM_) {
            for (int L = 0; L < 2; ++L) {
                int b = 4 + 10 * L;
                Wq[L] = F(b + 0); bq[L] = F(b + 1); Wk[L] = F(b + 2); bk[L] = F(b + 3);
                Wv[L] = F(b + 4); bv[L] = F(b + 5); We[L] = F(b + 6); Ws[L] = F(b + 7);
                bs[L] = F(b + 8); Wb[L] = F(b + 9);
            }
            for (int L = 0; L < 2; ++L) {
                int b = 24 + 4 * L;
                gW1[L] = F(b); gb1[L] = F(b + 1); gW2[L] = F(b + 2); gb2[L] = F(b + 3);
            }
            roW1 = F(32); rob1 = F(33); roW2 = F(34); rob2 = F(35);
        } else {
            load_sorted_params(4);
        }
    } else {
        // fully sorted pytree: batch_index, edge_attr, edge_index, params..., x
        bidx = I(0); ea = F(1); eidx = I(2);
        int p = load_sorted_params(3);
        x = F(p);
    }

    const int N = NN_, E = EE_, B = out_size;  // T == 1
    const int* src = eidx;
    const int* dst = eidx + E;

    // ---- workspace carve-up (~227 MB of f32) -----------------------------
    float* w = (float*)d_ws;
    float* q_   = w; w += (size_t)N * HDIM_;
    float* kf   = w; w += (size_t)N * HDIM_;
    float* vf   = w; w += (size_t)N * HDIM_;
    float* skp  = w; w += (size_t)N * HDIM_;
    float* outb = w; w += (size_t)N * HDIM_;
    float* h0   = w; w += (size_t)N * HDIM_;
    float* h1   = w; w += (size_t)N * HDIM_;
    float* agg  = w; w += (size_t)N * HDIM_;
    float* lg   = w; w += (size_t)E * HH_;
    float* sb   = w; w += (size_t)N * HH_;
    unsigned* me = (unsigned*)w; w += (size_t)N * HH_;
    float* gt   = w; w += (size_t)B * HDIM_;
    float* cnt  = w; w += (size_t)B;
    float* gpl  = w; w += (size_t)B * DD_;
    float* gh   = w; w += (size_t)N * DD_;
    float* gh2  = w; w += (size_t)N * DD_;
    float* zz   = w; w += (size_t)B * (HDIM_ + DD_);
    float* r1   = w; w += (size_t)B * DD_;

    auto blk = [](size_t n) { return (unsigned)((n + 255) / 256); };
    auto fillf = [&](float* p, float v, size_t n) {
        fill_f32<<<blk(n), 256, 0, stream>>>(p, v, n);
    };
    auto gemm = [&](const float* X, const float* W_, const float* B_, float* Y,
                    int M, int K, int Nc, int relu) {
        // fast path requires M%16==0, K%16==0, Nc%32==0 (true for all calls)
        int tiles = (M >> 4) * (Nc >> 5);
        gemm_wmma_f32<<<(tiles + 7) / 8, 256, 0, stream>>>(X, W_, B_, Y, M, K, Nc, relu);
    };

    // ---- 2x TransformerConv ---------------------------------------------
    auto translayer = [&](const float* hin, float* hout, int L) {
        gemm(hin, Wq[L], bq[L], q_,  N, HDIM_, HDIM_, 0);
        gemm(hin, Wk[L], bk[L], kf,  N, HDIM_, HDIM_, 0);
        gemm(hin, Wv[L], bv[L], vf,  N, HDIM_, HDIM_, 0);
        gemm(hin, Ws[L], bs[L], skp, N, HDIM_, HDIM_, 0);
        fill_u32<<<blk((size_t)N * HH_), 256, 0, stream>>>(me, 0u, (size_t)N * HH_);
        fillf(sb, 0.f, (size_t)N * HH_);
        fillf(outb, 0.f, (size_t)N * HDIM_);
        const size_t tE = (size_t)E * HH_;
        edge_logits <<<blk(tE), 256, 0, stream>>>(q_, kf, ea, We[L], src, dst, lg, me, E);
        edge_softmax<<<blk(tE), 256, 0, stream>>>(lg, me, dst, sb, E);
        edge_message<<<blk(tE), 256, 0, stream>>>(vf, ea, We[L], lg, sb, src, dst, outb, E);
        beta_combine<<<(N + 7) / 8, 256, 0, stream>>>(outb, skp, Wb[L], hout, N);
    };
    translayer(x,  h0, 0);
    translayer(h0, h1, 1);

    // ---- global mean pool of transformer embeddings ----------------------
    fillf(gt, 0.f, (size_t)B * HDIM_);
    fillf(cnt, 0.f, (size_t)B);
    pool_accum<<<blk((size_t)N * HDIM_), 256, 0, stream>>>(h1, bidx, gt, cnt, N, 7);

    // ---- GIN layer 1 (C=128 -> 32 -> 32) ---------------------------------
    copy_f32<<<blk((size_t)N * HDIM_), 256, 0, stream>>>(agg, h1, (size_t)N * HDIM_);
    gin_scatter<<<blk((size_t)E * HDIM_), 256, 0, stream>>>(h1, src, dst, agg, E, 7);
    gemm(agg, gW1[0], gb1[0], gh,  N, HDIM_, DD_, 1);
    gemm(gh,  gW2[0], gb2[0], gh2, N, DD_,   DD_, 1);

    // ---- GIN layer 2 (C=32) ----------------------------------------------
    copy_f32<<<blk((size_t)N * DD_), 256, 0, stream>>>(agg, gh2, (size_t)N * DD_);
    gin_scatter<<<blk((size_t)E * DD_), 256, 0, stream>>>(gh2, src, dst, agg, E, 5);
    gemm(agg, gW1[1], gb1[1], gh,  N, DD_, DD_, 1);
    gemm(gh,  gW2[1], gb2[1], gh2, N, DD_, DD_, 1);

    // ---- global add pool + readout ---------------------------------------
    fillf(gpl, 0.f, (size_t)B * DD_);
    pool_accum<<<blk((size_t)N * DD_), 256, 0, stream>>>(gh2, bidx, gpl, nullptr, N, 5);
    assemble_z<<<blk((size_t)B * (HDIM_ + DD_)), 256, 0, stream>>>(gt, cnt, gpl, zz, B);
    gemm(zz, roW1, rob1, r1, B, HDIM_ + DD_, DD_, 1);
    rowvec_dot<<<(B + 7) / 8, 256, 0, stream>>>(r1, roW2, rob2, (float*)d_out, B, DD_);
}